// GCNLayer_43688407335088
// MI455X (gfx1250) — compile-verified
//
#include <hip/hip_runtime.h>
#include <hip/hip_bf16.h>

typedef float v2f __attribute__((ext_vector_type(2)));
typedef float v8f __attribute__((ext_vector_type(8)));

#define IN_DIM  256
#define OUT_DIM 256
#define NHOP    3

// ---------------------------------------------------------------------------
// 1) in-degree via float atomics
__global__ __launch_bounds__(256) void deg_kernel(const int* __restrict__ dst,
                                                  float* __restrict__ deg, int nE) {
    int e = blockIdx.x * blockDim.x + threadIdx.x;
    if (e < nE) atomicAdd(&deg[dst[e]], 1.0f);
}

// 2) deg -> 1/max(deg,1) in place
__global__ __launch_bounds__(256) void dinv_kernel(float* __restrict__ deg, int n) {
    int i = blockIdx.x * blockDim.x + threadIdx.x;
    if (i < n) {
        float d = deg[i];
        deg[i] = 1.0f / (d > 1.0f ? d : 1.0f);
    }
}

// 3) edge scatter: agg[dst] += feat[src] * w   (64 threads per edge, float4 each)
__global__ __launch_bounds__(256) void scatter_kernel(const float* __restrict__ feat,
                                                      const int* __restrict__ src,
                                                      const int* __restrict__ dst,
                                                      const float* __restrict__ w,
                                                      float* __restrict__ agg, int nE) {
    long idx = (long)blockIdx.x * blockDim.x + threadIdx.x;
    int e = (int)(idx >> 6);
    if (e >= nE) return;
    int c = ((int)idx & 63) << 2;
    int s = src[e], d = dst[e];
    float we = w[e];
    const float4 v = *(const float4*)(feat + (long)s * IN_DIM + c);
    float* o = agg + (long)d * IN_DIM + c;
    atomicAdd(o + 0, v.x * we);
    atomicAdd(o + 1, v.y * we);
    atomicAdd(o + 2, v.z * we);
    atomicAdd(o + 3, v.w * we);
}

// 4) h *= dinv[row]
__global__ __launch_bounds__(256) void scale_kernel(float* __restrict__ h,
                                                    const float* __restrict__ dinv, int n) {
    long idx = (long)blockIdx.x * blockDim.x + threadIdx.x;
    int row = (int)(idx >> 6);
    if (row >= n) return;
    int c = ((int)idx & 63) << 2;
    float s = dinv[row];
    float4* p = (float4*)(h + (long)row * IN_DIM + c);
    float4 v = *p;
    v.x *= s; v.y *= s; v.z *= s; v.w *= s;
    *p = v;
}

// ---------------------------------------------------------------------------
// 5) fused GEMM (fp32 WMMA 16x16x4) + bias + ReLU + per-channel sum/sumsq.
//    Writes un-normalized activations into out[:, hop*256 : hop*256+256].
//    grid = (N/16, 2, 3), block = 256 (8 waves, each wave owns one 16x16 tile).
//    All tiles launched are fully in-range (grid.x = N/16), so the epilogue
//    is branch-free straight-line stores.
__global__ __launch_bounds__(256) void gemm_wmma_kernel(const float* __restrict__ x,
                                                        const float* __restrict__ h1,
                                                        const float* __restrict__ h2,
                                                        const float* __restrict__ Wmat,
                                                        const float* __restrict__ bias,
                                                        float* __restrict__ out,
                                                        float* __restrict__ stats,
                                                        int nNodes) {
    const int hop  = blockIdx.z;
    const float* ft = (hop == 0) ? x : (hop == 1) ? h1 : h2;
    const float* Wi = Wmat + (long)hop * IN_DIM * OUT_DIM;
    const float* bi = bias + hop * OUT_DIM;

    const int wave = threadIdx.x >> 5;
    const int lane = threadIdx.x & 31;
    const int col  = lane & 15;   // A: M index within tile; B/C: N index within tile
    const int half = lane >> 4;   // selects K pair / C row half

    const int m0 = blockIdx.x * 16;
    const int n0 = (blockIdx.y * 8 + wave) * 16;

    // A fragment source: row (m0+col), K offset half*2
    const float* arow = ft + (long)(m0 + col) * IN_DIM + half * 2;
    // B fragment source: column (n0+col), K rows (k + half*2 .. +1), row-major W [K][N]
    const float* bcol = Wi + n0 + col + (long)(half * 2) * OUT_DIM;

    v8f c = {};
#pragma unroll 8
    for (int k = 0; k < IN_DIM; k += 4) {
        v2f a; a.x = arow[k];                   a.y = arow[k + 1];
        v2f bb; bb.x = bcol[(long)k * OUT_DIM]; bb.y = bcol[(long)(k + 1) * OUT_DIM];
        c = __builtin_amdgcn_wmma_f32_16x16x4_f32(
                /*neg_a=*/false, a, /*neg_b=*/false, bb,
                /*c_mod=*/(short)0, c, /*reuse_a=*/false, /*reuse_b=*/false);
    }

    // Epilogue: C VGPR j holds (M = m0 + half*8 + j, N = n0 + col)
    const float bv = bi[n0 + col];
    const int outcol = hop * OUT_DIM + n0 + col;
    float* orow = out + (long)(m0 + half * 8) * (NHOP * OUT_DIM) + outcol;
    float psum = 0.0f, psq = 0.0f;
#pragma unroll
    for (int j = 0; j < 8; ++j) {
        float v = c[j] + bv;
        v = v > 0.0f ? v : 0.0f;              // ReLU
        orow[(long)j * (NHOP * OUT_DIM)] = v; // branch-free store
        psum += v;
        psq  += v * v;
    }
    // lanes L and L+16 target the same channel: fold halves first, then one
    // atomic per channel per tile (half the atomics, zero same-address clash)
    psum += __shfl_xor(psum, 16, 32);
    psq  += __shfl_xor(psq, 16, 32);
    if (half == 0) {
        atomicAdd(&stats[hop * 2 * OUT_DIM + n0 + col], psum);
        atomicAdd(&stats[hop * 2 * OUT_DIM + OUT_DIM + n0 + col], psq);
    }
}

// 6) per-channel (scale, shift) from accumulated stats
__global__ __launch_bounds__(256) void params_kernel(const float* __restrict__ stats,
                                                     const float* __restrict__ gamma,
                                                     const float* __restrict__ beta,
                                                     float* __restrict__ params, int nNodes) {
    int i = blockIdx.x * blockDim.x + threadIdx.x;
    if (i >= NHOP * OUT_DIM) return;
    int hop = i >> 8, ch = i & (OUT_DIM - 1);
    float s  = stats[hop * 2 * OUT_DIM + ch];
    float sq = stats[hop * 2 * OUT_DIM + OUT_DIM + ch];
    float invN = 1.0f / (float)nNodes;
    float mean = s * invN;
    float var  = sq * invN - mean * mean;
    float scale = gamma[i] * rsqrtf(var + 1e-5f);
    params[i] = scale;
    params[NHOP * OUT_DIM + i] = beta[i] - mean * scale;
}

// 7) in-place normalization of d_out (float4 over N x 768)
__global__ __launch_bounds__(256) void norm_kernel(float* __restrict__ out,
                                                   const float* __restrict__ params, int n) {
    long idx = (long)blockIdx.x * blockDim.x + threadIdx.x;
    int row = (int)(idx / 192);
    if (row >= n) return;
    int c = (int)(idx % 192) * 4;
    float4* p = (float4*)(out + (long)row * (NHOP * OUT_DIM) + c);
    float4 v = *p;
    const float4 sc = *(const float4*)(params + c);
    const float4 sh = *(const float4*)(params + NHOP * OUT_DIM + c);
    v.x = v.x * sc.x + sh.x;
    v.y = v.y * sc.y + sh.y;
    v.z = v.z * sc.z + sh.z;
    v.w = v.w * sc.w + sh.w;
    *p = v;
}

// ---------------------------------------------------------------------------
extern "C" void kernel_launch(void* const* d_in, const int* in_sizes, int n_in,
                              void* d_out, int out_size, void* d_ws, size_t ws_size,
                              hipStream_t stream) {
    const float* x     = (const float*)d_in[0];
    const float* w     = (const float*)d_in[1];
    const float* Wmat  = (const float*)d_in[2];
    const float* bias  = (const float*)d_in[3];
    const float* gamma = (const float*)d_in[4];
    const float* beta  = (const float*)d_in[5];
    const int*   src   = (const int*)d_in[6];
    const int*   dst   = (const int*)d_in[7];
    float* out = (float*)d_out;

    const int N = in_sizes[0] / IN_DIM;   // 50000
    const int E = in_sizes[1];            // 800000

    // workspace layout (floats)
    float* ws = (float*)d_ws;
    size_t o = 0;
    float* deg    = ws + o; o += (size_t)((N + 63) & ~63);
    float* stats  = ws + o; o += NHOP * 2 * OUT_DIM;   // 1536
    float* params = ws + o; o += NHOP * 2 * OUT_DIM;   // 1536
    o = (o + 63) & ~(size_t)63;
    float* h1 = ws + o; o += (size_t)N * IN_DIM;
    float* h2 = ws + o; o += (size_t)N * IN_DIM;

    // zero the accumulation buffers (atomics build into them each launch)
    hipMemsetAsync(deg,   0, (size_t)N * sizeof(float), stream);
    hipMemsetAsync(stats, 0, NHOP * 2 * OUT_DIM * sizeof(float), stream);
    hipMemsetAsync(h1,    0, (size_t)N * IN_DIM * sizeof(float), stream);
    hipMemsetAsync(h2,    0, (size_t)N * IN_DIM * sizeof(float), stream);

    // degree -> 1/deg
    deg_kernel <<<(E + 255) / 256, 256, 0, stream>>>(dst, deg, E);
    dinv_kernel<<<(N + 255) / 256, 256, 0, stream>>>(deg, N);

    // hop 1: h1 = D_in * (scatter x)
    long nscat = ((long)E * 64 + 255) / 256;
    long nscal = ((long)N * 64 + 255) / 256;
    scatter_kernel<<<(int)nscat, 256, 0, stream>>>(x, src, dst, w, h1, E);
    scale_kernel  <<<(int)nscal, 256, 0, stream>>>(h1, deg, N);

    // hop 2: h2 = D_in * (scatter h1)
    scatter_kernel<<<(int)nscat, 256, 0, stream>>>(h1, src, dst, w, h2, E);
    scale_kernel  <<<(int)nscal, 256, 0, stream>>>(h2, deg, N);

    // fused fp32-WMMA GEMM + bias + ReLU + stats, all three hops
    dim3 ggrid(N / 16, 2, NHOP);
    gemm_wmma_kernel<<<ggrid, 256, 0, stream>>>(x, h1, h2, Wmat, bias, out, stats, N);

    // batch-norm finalize + in-place normalize of d_out
    params_kernel<<<NHOP, 256, 0, stream>>>(stats, gamma, beta, params, N);
    long nnorm = ((long)N * 192 + 255) / 256;
    norm_kernel<<<(int)nnorm, 256, 0, stream>>>(out, params, N);
}